// EdgeAttrGNNLight_79731772883553
// MI455X (gfx1250) — compile-verified
//
#include <hip/hip_runtime.h>
#include <hip/hip_bf16.h>

// ---------------------------------------------------------------------------
// EdgeAttrGNNLight on MI455X (gfx1250, wave32, WMMA)
//
// Pipeline:
//   Z: zero workspace (agg, cnt, gsum, gcnt)
//   B: edge kernel  -- per-16-edge tile, nn_w2(+b2) resident in 279KB LDS,
//                      chained V_WMMA_F32_16X16X4_F32 builds W_e chunks which
//                      are immediately contracted with x[src]; atomic mean-agg.
//   C: node kernel  -- x@root via WMMA, + agg/cnt + bias, ReLU, atomic pool.
//   D: head kernel  -- pooled @ lin_w + lin_b.
// ---------------------------------------------------------------------------

typedef __attribute__((ext_vector_type(2))) float v2f;
typedef __attribute__((ext_vector_type(8))) float v8f;

constexpr int N_NODES   = 50000;
constexpr int N_EDGES   = 200000;
constexpr int IN_F      = 32;
constexpr int HID       = 32;
constexpr int N_CLASSES = 2;
constexpr int N_GRAPHS  = 64;

// LDS layout for edge kernel: W2T is nn_w2 transposed to [column][j] with a
// stride of 68 floats. 68 = 64 (K rows) + 4 pad; since 68 ≡ 4 (mod 64), the
// per-lane bank index of a B-fragment ds_load_b64 is 4*lane + (2g+d): all 64
// DWORDs of a wave's load hit 64 distinct banks (conflict-free). Row j=64
// carries nn_b2 (bias folded into the GEMM via an augmented H column == 1),
// rows 65..67 are zero.
constexpr int W2T_STRIDE = 68;
constexpr int SMEM_EDGE_FLOATS = 1024 * W2T_STRIDE + 64 + 64;

__global__ void egnn_zero_kernel(float* __restrict__ p, int n) {
  int i = blockIdx.x * blockDim.x + threadIdx.x;
  if (i < n) p[i] = 0.0f;
}

__global__ void egnn_edge_kernel(const float* __restrict__ x,
                                 const int*   __restrict__ ei,
                                 const float* __restrict__ ea,
                                 const float* __restrict__ w1,
                                 const float* __restrict__ b1,
                                 const float* __restrict__ w2,
                                 const float* __restrict__ b2,
                                 float* __restrict__ agg,
                                 float* __restrict__ cnt) {
  extern __shared__ float smem[];
  float* W2T = smem;                       // 1024 * 68
  float* w1s = smem + 1024 * W2T_STRIDE;   // 64
  float* b1s = w1s + 64;                   // 64

  const int tid = threadIdx.x;

  // Stage nn_w2 transposed into LDS: W2T[c*68 + j] = w2[j*1024 + c]
  for (int idx = tid; idx < 64 * 1024; idx += blockDim.x) {
    int j = idx >> 10;
    int c = idx & 1023;
    W2T[c * W2T_STRIDE + j] = w2[idx];
  }
  // Bias row (j = 64) + zero padding rows.
  for (int c = tid; c < 1024; c += blockDim.x) {
    W2T[c * W2T_STRIDE + 64] = b2[c];
    W2T[c * W2T_STRIDE + 65] = 0.0f;
    W2T[c * W2T_STRIDE + 66] = 0.0f;
    W2T[c * W2T_STRIDE + 67] = 0.0f;
  }
  if (tid < 64) { w1s[tid] = w1[tid]; b1s[tid] = b1[tid]; }
  __syncthreads();

  const int wave = tid >> 5;
  const int lane = tid & 31;
  const int g    = lane >> 4;   // half-group: lanes 16..31 handle M rows +8
  const int n16  = lane & 15;   // N (output hid column within 16-col half)

  const int nTiles  = N_EDGES / 16;       // 12500
  const int wavesWG = blockDim.x >> 5;
  int tile   = blockIdx.x * wavesWG + wave;
  int stride = gridDim.x * wavesWG;

  for (; tile < nTiles; tile += stride) {
    const int base = tile * 16;

    // Edge metadata for the M rows this lane owns (e = v + 8*g).
    int srcv[8], dstv[8];
#pragma unroll
    for (int v = 0; v < 8; ++v) {
      srcv[v] = ei[base + v + 8 * g];
      dstv[v] = ei[N_EDGES + base + v + 8 * g];
    }

    // A fragments: H'[e, j] = relu(a_e*w1[j] + b1[j]) for j<64, 1 at j==64.
    // A layout (16x4 f32): lane l holds M=l%16, VGPR0 -> K=2g, VGPR1 -> K=2g+1.
    const float aE = ea[base + n16];
    v2f hA[17];
#pragma unroll
    for (int kk = 0; kk < 16; ++kk) {
      const int j0 = 4 * kk + 2 * g;
      v2f wp = *(const v2f*)&w1s[j0];
      v2f bp = *(const v2f*)&b1s[j0];
      hA[kk].x = fmaxf(aE * wp.x + bp.x, 0.0f);
      hA[kk].y = fmaxf(aE * wp.y + bp.y, 0.0f);
    }
    hA[16].x = (g == 0) ? 1.0f : 0.0f;  // augmented column for bias (j=64)
    hA[16].y = 0.0f;                    // j=65..67 are zero

    v8f m0 = {};  // msg[e, n16]
    v8f m1 = {};  // msg[e, n16+16]

    for (int i = 0; i < IN_F; ++i) {
      // x[src_e, i]; 16-lane broadcast loads, L1-resident.
      float xv[8];
#pragma unroll
      for (int v = 0; v < 8; ++v) xv[v] = x[srcv[v] * IN_F + i];

      const float* B0 = &W2T[(i * 32 + n16)      * W2T_STRIDE + 2 * g];
      const float* B1 = &W2T[(i * 32 + 16 + n16) * W2T_STRIDE + 2 * g];

      v8f acc0 = {};
      v8f acc1 = {};
#pragma unroll
      for (int kk = 0; kk < 17; ++kk) {
        v2f bf0 = *(const v2f*)(B0 + 4 * kk);  // conflict-free ds_load_b64
        v2f bf1 = *(const v2f*)(B1 + 4 * kk);
        acc0 = __builtin_amdgcn_wmma_f32_16x16x4_f32(
            false, hA[kk], false, bf0, (short)0, acc0, false, false);
        acc1 = __builtin_amdgcn_wmma_f32_16x16x4_f32(
            false, hA[kk], false, bf1, (short)0, acc1, false, false);
      }
      // acc = W_e[i, h] chunk (bias included); contract with x[src,i].
#pragma unroll
      for (int v = 0; v < 8; ++v) {
        m0[v] += xv[v] * acc0[v];
        m1[v] += xv[v] * acc1[v];
      }
    }

    // Scatter-add messages into per-dst accumulator (mean numerator).
#pragma unroll
    for (int v = 0; v < 8; ++v) {
      atomicAdd(&agg[dstv[v] * HID + n16],      m0[v]);
      atomicAdd(&agg[dstv[v] * HID + n16 + 16], m1[v]);
    }
    if (g == 0) atomicAdd(&cnt[ei[N_EDGES + base + n16]], 1.0f);
  }
}

__global__ void egnn_node_pool_kernel(const float* __restrict__ x,
                                      const int*   __restrict__ batch,
                                      const float* __restrict__ root,
                                      const float* __restrict__ conv_bias,
                                      const float* __restrict__ agg,
                                      const float* __restrict__ cnt,
                                      float* __restrict__ gsum,
                                      float* __restrict__ gcnt) {
  // rootT[h*36 + i] = root[i*32 + h]; stride 36 = 4*9, 9 odd -> B-fragment
  // ds_load_b64 is bank-conflict free (4*(9n mod 16) + (2g+d) all distinct).
  __shared__ float rootT[HID * 36];
  const int tid = threadIdx.x;
  for (int idx = tid; idx < IN_F * HID; idx += blockDim.x) {
    int i = idx >> 5;
    int h = idx & 31;
    rootT[h * 36 + i] = root[idx];
  }
  __syncthreads();

  const int wave = tid >> 5;
  const int lane = tid & 31;
  const int g    = lane >> 4;
  const int n16  = lane & 15;

  const int tile = blockIdx.x * (blockDim.x >> 5) + wave;
  if (tile >= N_NODES / 16) return;
  const int t0  = tile * 16;
  const int row = t0 + n16;  // A-matrix M row = node

  v8f acc0 = {};
  v8f acc1 = {};
#pragma unroll
  for (int kk = 0; kk < 8; ++kk) {  // K = 32 input features
    v2f a  = *(const v2f*)&x[row * IN_F + 4 * kk + 2 * g];
    v2f b0 = *(const v2f*)&rootT[n16 * 36        + 4 * kk + 2 * g];
    v2f b1 = *(const v2f*)&rootT[(n16 + 16) * 36 + 4 * kk + 2 * g];
    acc0 = __builtin_amdgcn_wmma_f32_16x16x4_f32(
        false, a, false, b0, (short)0, acc0, false, false);
    acc1 = __builtin_amdgcn_wmma_f32_16x16x4_f32(
        false, a, false, b1, (short)0, acc1, false, false);
  }

  const float cb0 = conv_bias[n16];
  const float cb1 = conv_bias[n16 + 16];
#pragma unroll
  for (int v = 0; v < 8; ++v) {
    const int n = t0 + v + 8 * g;
    const float inv = 1.0f / fmaxf(cnt[n], 1.0f);
    const int b = batch[n];
    float val0 = fmaxf(acc0[v] + agg[n * HID + n16]      * inv + cb0, 0.0f);
    float val1 = fmaxf(acc1[v] + agg[n * HID + n16 + 16] * inv + cb1, 0.0f);
    atomicAdd(&gsum[b * HID + n16],      val0);
    atomicAdd(&gsum[b * HID + n16 + 16], val1);
    if (n16 == 0) atomicAdd(&gcnt[b], 1.0f);  // one count per node
  }
}

__global__ void egnn_head_kernel(const float* __restrict__ gsum,
                                 const float* __restrict__ gcnt,
                                 const float* __restrict__ lin_w,
                                 const float* __restrict__ lin_b,
                                 float* __restrict__ out) {
  const int t = blockIdx.x * blockDim.x + threadIdx.x;
  if (t >= N_GRAPHS * N_CLASSES) return;
  const int gph = t / N_CLASSES;
  const int c   = t % N_CLASSES;
  const float inv = 1.0f / fmaxf(gcnt[gph], 1.0f);
  float s = lin_b[c];
#pragma unroll
  for (int h = 0; h < HID; ++h)
    s += gsum[gph * HID + h] * inv * lin_w[h * N_CLASSES + c];
  out[t] = s;
}

extern "C" void kernel_launch(void* const* d_in, const int* in_sizes, int n_in,
                              void* d_out, int out_size, void* d_ws, size_t ws_size,
                              hipStream_t stream) {
  const float* x     = (const float*)d_in[0];
  const int*   ei    = (const int*)  d_in[1];   // [2, E]
  const float* ea    = (const float*)d_in[2];   // [E, 1]
  const int*   batch = (const int*)  d_in[3];
  const float* w1    = (const float*)d_in[4];   // [1, 64]
  const float* b1    = (const float*)d_in[5];   // [64]
  const float* w2    = (const float*)d_in[6];   // [64, 1024]
  const float* b2    = (const float*)d_in[7];   // [1024]
  const float* root  = (const float*)d_in[8];   // [32, 32]
  const float* cbias = (const float*)d_in[9];   // [32]
  const float* lw    = (const float*)d_in[10];  // [32, 2]
  const float* lb    = (const float*)d_in[11];  // [2]
  float* out = (float*)d_out;

  // Workspace: agg [N,32] | cnt [N] | gsum [G,32] | gcnt [G]
  float* agg  = (float*)d_ws;
  float* cnt  = agg + (size_t)N_NODES * HID;
  float* gsum = cnt + N_NODES;
  float* gcnt = gsum + N_GRAPHS * HID;
  const int zeroN = N_NODES * HID + N_NODES + N_GRAPHS * HID + N_GRAPHS;

  egnn_zero_kernel<<<(zeroN + 255) / 256, 256, 0, stream>>>(agg, zeroN);

  const size_t smemEdge = (size_t)SMEM_EDGE_FLOATS * sizeof(float);  // ~279 KB
  (void)hipFuncSetAttribute((const void*)egnn_edge_kernel,
                            hipFuncAttributeMaxDynamicSharedMemorySize,
                            (int)smemEdge);
  egnn_edge_kernel<<<256, 256, smemEdge, stream>>>(x, ei, ea, w1, b1, w2, b2,
                                                   agg, cnt);

  const int nodeTiles = N_NODES / 16;  // 3125
  egnn_node_pool_kernel<<<(nodeTiles + 7) / 8, 256, 0, stream>>>(
      x, batch, root, cbias, agg, cnt, gsum, gcnt);

  egnn_head_kernel<<<1, 128, 0, stream>>>(gsum, gcnt, lw, lb, out);
}